// BeliefPlausibilityFocused_35656818492191
// MI455X (gfx1250) — compile-verified
//
#include <hip/hip_runtime.h>

// Belief/Plausibility broadcast-fill for MI455X (gfx1250).
// Outputs are pure channel-index masks (inputs' values are never read):
//   bel[..., j] = ((j & focal) == focal),  pl[..., j] = ((j & focal) != 0)
// => 245 MB of pattern stores, ~10.5 us floor @ 23.3 TB/s. Store-bound.
// Primary path: CDNA5 async LDS->global 128-bit stores (ASYNCcnt-tracked);
// pattern staged once in LDS per workgroup. Exact-cover launch: one thread
// per 16-byte vector pair, no inner loop.

typedef float v4f __attribute__((ext_vector_type(4)));
typedef int   v4i __attribute__((ext_vector_type(4)));

typedef __attribute__((address_space(1))) v4i* gp_v4i;
typedef __attribute__((address_space(3))) v4i* lp_v4i;

#if defined(__has_builtin)
#  if __has_builtin(__builtin_amdgcn_global_store_async_from_lds_b128)
#    define USE_ASYNC_STORE 1
#  else
#    define USE_ASYNC_STORE 0
#  endif
#  if __has_builtin(__builtin_amdgcn_s_wait_asynccnt)
#    define HAVE_WAIT_ASYNC_BUILTIN 1
#  else
#    define HAVE_WAIT_ASYNC_BUILTIN 0
#  endif
#else
#  define USE_ASYNC_STORE 0
#  define HAVE_WAIT_ASYNC_BUILTIN 0
#endif

__global__ __launch_bounds__(256) void bel_pl_fill_kernel(
    float* __restrict__ out,          // [bel (N floats) | pl (N floats)]
    const int* __restrict__ focal_p,  // device scalar
    long long N,                      // elements per output tensor
    int n_sets)                       // power of two, channel count
{
    // Pattern staged in LDS, replicated slightly past n_sets so any aligned
    // 16-byte window starting at (e & (n_sets-1)) is valid.
    __shared__ __align__(16) float pat_bel[544];
    __shared__ __align__(16) float pat_pl[544];

    const int focal = *focal_p;
    const int mask  = n_sets - 1;

    int L = (n_sets < 4 ? 4 : n_sets) + 4;
    if (L > 544) L = 544;
    for (int i = threadIdx.x; i < L; i += blockDim.x) {
        const int j = i & mask;
        const int contain = j & focal;
        pat_bel[i] = (contain == focal) ? 1.0f : 0.0f;
        pat_pl[i]  = (contain != 0)     ? 1.0f : 0.0f;
    }
    __syncthreads();

    const long long nvec = N >> 2;                       // 16B vectors per output
    const long long tid  = (long long)blockIdx.x * 256 + threadIdx.x;

    if (tid < nvec) {
        const long long e = tid << 2;
        const int off = ((int)e) & mask;                 // multiple of 4 (or 0)
#if USE_ASYNC_STORE
        __builtin_amdgcn_global_store_async_from_lds_b128(
            (gp_v4i)(out + e), (lp_v4i)(pat_bel + off), 0, 0);
        __builtin_amdgcn_global_store_async_from_lds_b128(
            (gp_v4i)(out + N + e), (lp_v4i)(pat_pl + off), 0, 0);
#else
        const v4f b = *(const v4f*)(pat_bel + off);
        const v4f p = *(const v4f*)(pat_pl + off);
        __builtin_nontemporal_store(b, (v4f*)(out + e));
        __builtin_nontemporal_store(p, (v4f*)(out + N + e));
#endif
    }

    // Scalar tail for N % 4 (zero for this shape; kept for generality).
    const long long tail = N & 3;
    if (tid < tail) {
        const long long e = (nvec << 2) + tid;
        const int j = (int)(e & (long long)mask);
        const int contain = j & focal;
        out[e]     = (contain == focal) ? 1.0f : 0.0f;
        out[N + e] = (contain != 0)     ? 1.0f : 0.0f;
    }

#if USE_ASYNC_STORE
#  if HAVE_WAIT_ASYNC_BUILTIN
    __builtin_amdgcn_s_wait_asynccnt(0);
#  else
    asm volatile("s_wait_asynccnt 0" ::: "memory");
#  endif
#endif
}

extern "C" void kernel_launch(void* const* d_in, const int* in_sizes, int n_in,
                              void* d_out, int out_size, void* d_ws, size_t ws_size,
                              hipStream_t stream) {
    (void)d_ws; (void)ws_size; (void)n_in;
    const int* focal_p = (const int*)d_in[1];   // python scalar -> 1-elem int array
    float* out = (float*)d_out;

    const long long in0 = (long long)in_sizes[0];     // B*H*W*C
    const long long N   = (long long)out_size / 2;    // elements per output

    // Recover C (last dim) from: out_size == 2 * (in0/C) * 2^(C-1)
    //  <=> out_size * C == in0 * 2^C.  For this problem: C=4 -> n_sets=8.
    int n_sets = 8;
    for (int C = 1; C <= 30; ++C) {
        if ((long long)out_size * C == in0 * (1LL << C)) {
            n_sets = 1 << (C - 1);
            break;
        }
    }

    const int block = 256;
    const long long nvec = N >> 2;
    long long grid = (nvec + block - 1) / block;        // exact cover, no loop
    if (grid < 1) grid = 1;

    bel_pl_fill_kernel<<<(int)grid, block, 0, stream>>>(out, focal_p, N, n_sets);
}